// CustomDistribution_6837587935978
// MI455X (gfx1250) — compile-verified
//
#include <hip/hip_runtime.h>
#include <hip/hip_bf16.h>

// ---------------------------------------------------------------------------
// Tanh-Normal grid sampler for MI455X (gfx1250, wave32).
//
//   probs_raw[e][i] = exp2( a2[e]*g2[i] + b2[e]*g1[i] + c2[e]*1 + 1*lj2[i] )
//     g1 = atanh(grid), g2 = g1^2, lj2 = -log2(1-grid^2)
//     a2 = -log2e/(2s^2), b2 = log2e*mu/s^2, c2 = -log2e*mu^2/(2s^2)
//
// The full exponent (including Jacobian) is a rank-4 product computed by
// V_WMMA_F32_16X16X4_F32 with C = inline 0:
//     A(16x4) = [a2 b2 c2 1] per element-row, B(4x16) = [g2 g1 1 lj2] per col.
// Its D tile feeds v_exp_f32 (= 2^x) directly -- no VALU between matrix pipe
// and trans pipe. Per-lane partials stay in D-layout for 5 chunks before one
// ds_swizzle tree reduction -> 25 cumulative checkpoints; the inverse-CDF
// search scans those and re-evaluates <=80 points.
// ---------------------------------------------------------------------------

typedef __attribute__((ext_vector_type(2))) float v2f;
typedef __attribute__((ext_vector_type(8))) float v8f;

#define EPS_F        1.1920928955078125e-7f
#define NSAMP        2000
#define NCHUNK       125           // 2000 / 16 grid points per WMMA tile
#define GROUP        5             // chunks per checkpoint
#define NCKPT        (NCHUNK / GROUP)          // 25 checkpoints
#define PTS_PER_CKPT (GROUP * 16)              // 80 grid points
#define Y0_F         0.9999f
#define STEP_F       (2.0f * Y0_F / (float)(NSAMP - 1))
#define WAVES_PB     4             // 4 waves * 16 elements = 64 elements/block
#define INV_SQRT2PI  0.3989422804014327f
#define SQRT2PI      2.5066282746310002f
#define LOG2E_F      1.4426950408889634f

#if __has_builtin(__builtin_amdgcn_wmma_f32_16x16x4_f32)
#define HAVE_WMMA_16X16X4 1
#else
#define HAVE_WMMA_16X16X4 0
#endif

// XOR-shuffle add across each 16-lane half (masks 1,2,4,8 stay in-half).
// ds_swizzle_b32 group-of-32 mode: offset = (xor_mask << 10) | 0x1f.
__device__ __forceinline__ float half16_sum(float v) {
    v += __int_as_float(__builtin_amdgcn_ds_swizzle(__float_as_int(v), 0x041f));
    v += __int_as_float(__builtin_amdgcn_ds_swizzle(__float_as_int(v), 0x081f));
    v += __int_as_float(__builtin_amdgcn_ds_swizzle(__float_as_int(v), 0x101f));
    v += __int_as_float(__builtin_amdgcn_ds_swizzle(__float_as_int(v), 0x201f));
    return v;
}

__device__ __forceinline__ float exp2f_fast(float x) {
#if __has_builtin(__builtin_amdgcn_exp2f)
    return __builtin_amdgcn_exp2f(x);
#else
    return exp2f(x);
#endif
}

__global__ __launch_bounds__(WAVES_PB * 32)
void tanh_normal_sample_kernel(const float* __restrict__ mean,
                               const float* __restrict__ std_in,
                               const float* __restrict__ uni,
                               float* __restrict__ out,
                               int n_elems)
{
    // Pre-swizzled B table {g2, g1, 1.0, lj2}: lo half loads .xy, hi half
    // loads .zw -- one ds_load_b64 each, no selects. 32 KB.
    __shared__ float4 tbl[NSAMP];
    // Cumulative checkpoint sums: chk[wave][ckpt][element]. 6.4 KB.
    __shared__ float chk[WAVES_PB][NCKPT][16];

    const int tid = threadIdx.x;

    // ---- build grid tables cooperatively (amortized 64x per block) -------
    for (int i = tid; i < NSAMP; i += blockDim.x) {
        float x    = fmaf((float)i, STEP_F, -Y0_F);
        float r    = __builtin_amdgcn_rcpf(1.0f - x);
        float at   = 0.5f * __logf(fmaf(1.0f + x, r, EPS_F));  // ref atanh formula
        float omx2 = fmaf(-x, x, 1.0f);                        // 1 - x^2
        tbl[i] = make_float4(at * at, at, 1.0f, -__log2f(omx2));
    }
    __syncthreads();

    const int wave  = tid >> 5;
    const int lane  = tid & 31;
    const int col16 = lane & 15;                 // tile column == element slot
    const bool hi   = lane >= 16;
    const int rb    = hi ? 8 : 0;                // D-layout row base for this half
    const int ebase = (blockIdx.x * WAVES_PB + wave) * 16;
    const bool valid = (ebase + col16) < n_elems;

    // per-element Gaussian coefficients, pre-scaled by log2(e) so the WMMA
    // output feeds v_exp_f32 (2^x) with no intervening multiply
    float mu = valid ? mean[ebase + col16]   : 0.0f;
    float sg = (valid ? std_in[ebase + col16] : 1.0f) + EPS_F;
    float inv_s2 = __builtin_amdgcn_rcpf(sg * sg);
    float a2 = -0.5f * LOG2E_F * inv_s2;                 // * g^2
    float b2 = LOG2E_F * mu * inv_s2;                    // * g
    float c2 = -0.5f * LOG2E_F * mu * mu * inv_s2;       // * 1

#if HAVE_WMMA_16X16X4
    // A(16x4) f32 layout: lanes 0-15 hold K=0,1 ; lanes 16-31 hold K=2,3.
    // Row = [a2, b2, c2, 1]: K=3 picks up the per-column Jacobian from B.
    v2f Aop;
    Aop.x = hi ? c2   : a2;
    Aop.y = hi ? 1.0f : b2;
#endif

    // per-half B fetch pointer: lo half reads {g2,g1}, hi half reads {1,lj2}
    const float2* tbl2 = (const float2*)tbl;

    // ---- pass 1: WMMA exponent tiles, exp2, grouped checkpoint reduce ----
    v8f cum = {};                                // reduced cumulative (mirrored)
    for (int g = 0; g < NCKPT; ++g) {
        v8f loc = {};                            // per-lane D-layout partials
        #pragma unroll
        for (int j = 0; j < GROUP; ++j) {
            const int n = (g * GROUP + j) * 16 + col16;
            // B(4x16): lanes 0-15 hold K=0,1 = (g2,g1); lanes 16-31 hold
            // K=2,3 = (1, lj2). One ds_load_b64, no cndmask.
            const float2 Bv = tbl2[2 * n + (hi ? 1 : 0)];

            v8f E;
#if HAVE_WMMA_16X16X4
            v2f Bop; Bop.x = Bv.x; Bop.y = Bv.y;
            v8f Czero = {};
            E = __builtin_amdgcn_wmma_f32_16x16x4_f32(false, Aop, false, Bop,
                                                      (short)0, Czero, false, false);
#else
            // VALU fallback: broadcast row coefficients and FMA.
            const float2 Blo = tbl2[2 * n];      // (g2, g1)
            const float2 Bhi = tbl2[2 * n + 1];  // (1, lj2)
            #pragma unroll
            for (int k = 0; k < 8; ++k) {
                int row  = rb + k;
                float ar = __shfl(a2, row, 32);
                float br = __shfl(b2, row, 32);
                float cr = __shfl(c2, row, 32);
                E[k] = fmaf(ar, Blo.x, fmaf(br, Blo.y, cr)) + Bhi.y;
            }
#endif
            // probs tile: 8 v_exp_f32 per lane (trans pipe), accumulate in-layout
            #pragma unroll
            for (int k = 0; k < 8; ++k) loc[k] += exp2f_fast(E[k]);
        }
        // one tree reduction per 80 grid points (DS pipe stays light)
        #pragma unroll
        for (int k = 0; k < 8; ++k) cum[k] += half16_sum(loc[k]);
        if (col16 == 0) {
            #pragma unroll
            for (int k = 0; k < 8; ++k) chk[wave][g][rb + k] = cum[k];
        }
    }
    __syncthreads();

    // ---- pass 2: inverse-CDF search via checkpoints (lanes 0-15) ---------
    if (lane < 16 && valid) {
        const int   e       = lane;
        const float u       = uni[ebase + e];
        const float S_raw   = chk[wave][NCKPT - 1][e];
        const float invnorm = INV_SQRT2PI * __builtin_amdgcn_rcpf(sg);
        const float S_ref   = invnorm * S_raw;
        // cdf[i] > u  <=>  cum_raw[i] > u*(S_ref+EPS)/invnorm
        const float T = u * (S_ref + EPS_F) * (sg * SQRT2PI);

        int c = 0;
        while (c < NCKPT && !(chk[wave][c][e] > T)) ++c;

        int   idx;
        float p_sel;
        if (c >= NCKPT) {
            // u beyond final cdf (rare): jnp.argmax(all-False) == 0
            const float4 t0 = tbl[0];
            idx   = 0;
            p_sel = exp2f_fast(fmaf(a2, t0.x, fmaf(b2, t0.y, c2)) + t0.w);
        } else {
            float run = (c == 0) ? 0.0f : chk[wave][c - 1][e];
            const int n0 = c * PTS_PER_CKPT;
            idx = n0 + PTS_PER_CKPT - 1;
            float p = 0.0f;
            #pragma unroll 1
            for (int i = 0; i < PTS_PER_CKPT; ++i) {
                const float4 t = tbl[n0 + i];
                p = exp2f_fast(fmaf(a2, t.x, fmaf(b2, t.y, c2)) + t.w);
                run += p;
                if (run > T) { idx = n0 + i; break; }
            }
            p_sel = p;
        }

        const float val  = fmaf((float)idx, STEP_F, -Y0_F);
        const float pout = invnorm * p_sel * __builtin_amdgcn_rcpf(S_ref + EPS_F);
        out[ebase + e]           = val;    // sampled_values
        out[n_elems + ebase + e] = pout;   // sampled_probs
    }
}

extern "C" void kernel_launch(void* const* d_in, const int* in_sizes, int n_in,
                              void* d_out, int out_size, void* d_ws, size_t ws_size,
                              hipStream_t stream)
{
    const float* mean = (const float*)d_in[0];
    const float* stdv = (const float*)d_in[1];
    const float* uni  = (const float*)d_in[2];
    float* out = (float*)d_out;

    const int n_elems = in_sizes[0];                       // 4096*16 = 65536
    const int elems_per_block = WAVES_PB * 16;             // 64
    const int blocks = (n_elems + elems_per_block - 1) / elems_per_block;

    hipLaunchKernelGGL(tanh_normal_sample_kernel,
                       dim3(blocks), dim3(WAVES_PB * 32), 0, stream,
                       mean, stdv, uni, out, n_elems);
}